// LocallyDense_42872363548771
// MI455X (gfx1250) — compile-verified
//
#include <hip/hip_runtime.h>

typedef __attribute__((ext_vector_type(2))) float v2f;
typedef __attribute__((ext_vector_type(8))) float v8f;
typedef int i32x4 __attribute__((vector_size(16)));   // matches builtin's V4i param type

#define AS1 __attribute__((address_space(1)))
#define AS3 __attribute__((address_space(3)))

// Problem constants (from reference setup_inputs)
constexpr int B = 256, D = 65536, NGRP = 41, G = 1024, E = 256, DEPTH = 2;
constexpr int BT = 64;           // batch rows per workgroup
constexpr int KC = 64;           // K chunk staged in LDS
constexpr float EPS = 1e-3f;

// LDS layout (in floats)
constexpr int HS = 258;                    // h buffer row stride (pad -> bank step 2)
constexpr int WS = 272;                    // weight chunk row stride (256+16)
constexpr int AS = 66;                     // gathered-A chunk row stride
constexpr int H0_OFF  = 0;                 // 16512 floats
constexpr int H1_OFF  = BT * HS;           // 16512
constexpr int WB0_OFF = 2 * BT * HS;       // 33024 (ping/pong stride KC*WS = 17408)
constexpr int GA0_OFF = 0;                 // gather ping/pong overlap h0 (dead then)
constexpr int LDS_FLOATS = WB0_OFF + 2 * KC * WS;   // 67840 floats = 271360 B < 320 KB

// ---------------- async global -> LDS copy helpers (CDNA5 ASYNCcnt path) ----------------
#if __has_builtin(__builtin_amdgcn_global_load_async_to_lds_b128) && \
    __has_builtin(__builtin_amdgcn_global_load_async_to_lds_b32)
#define HAVE_ASYNC_LDS 1
#else
#define HAVE_ASYNC_LDS 0
#endif

__device__ __forceinline__ void cp_b128(const float* g, float* l) {
#if HAVE_ASYNC_LDS
    __builtin_amdgcn_global_load_async_to_lds_b128(
        (AS1 i32x4*)g, (AS3 i32x4*)l, 0, 0);
#else
    *(float4*)l = *(const float4*)g;
#endif
}
__device__ __forceinline__ void cp_b32(const float* g, float* l) {
#if HAVE_ASYNC_LDS
    __builtin_amdgcn_global_load_async_to_lds_b32(
        (AS1 int*)g, (AS3 int*)l, 0, 0);
#else
    *l = *g;
#endif
}
__device__ __forceinline__ void wait_async_copies() {
#if HAVE_ASYNC_LDS
#if __has_builtin(__builtin_amdgcn_s_wait_asynccnt)
    __builtin_amdgcn_s_wait_asynccnt(0);
#else
    asm volatile("s_wait_asynccnt 0x0" ::: "memory");
#endif
#endif
}

__global__ __launch_bounds__(256, 1)
void locally_dense_kernel(const float* __restrict__ x,
                          const int*   __restrict__ gidx,
                          const float* __restrict__ W1,
                          const float* __restrict__ b1,
                          const float* __restrict__ gamma1, const float* __restrict__ beta1,
                          const float* __restrict__ mean1,  const float* __restrict__ var1,
                          const float* __restrict__ W2,
                          const float* __restrict__ b2,
                          const float* __restrict__ gamma2, const float* __restrict__ beta2,
                          const float* __restrict__ mean2,  const float* __restrict__ var2,
                          float* __restrict__ out)
{
    extern __shared__ float smem[];
    const int n    = blockIdx.x;           // group index 0..40
    const int b0   = blockIdx.y * BT;      // batch-tile origin
    const int tid  = threadIdx.x;
    const int lane = tid & 31;
    const int wave = tid >> 5;             // 0..7
    const int mtile = wave & 3;            // 16-row M tile within the 64-row B-tile
    const int nhalf = wave >> 2;           // 128-col half of E
    const int lrow  = lane & 15;
    const int lhi   = lane >> 4;           // K offset 0/2 in WMMA fragments
    const int arow  = mtile * 16 + lrow;   // A-fragment row
    const int wl_off = lhi * 2 * WS + nhalf * 128 + lrow;   // lane part of B-frag address
    const int al_off = arow * AS + lhi * 2;                 // lane part of layer-1 A address
    const int hl_off = arow * HS + lhi * 2;                 // lane part of layer-2/3 A address

    float* h0 = smem + H0_OFF;
    float* h1 = smem + H1_OFF;
    // ping/pong buffers selected by arithmetic (no pointer arrays -> no static addrspacecast)
    auto wb_ptr = [&](int idx) -> float* { return smem + WB0_OFF + idx * (KC * WS); };
    auto ga_ptr = [&](int idx) -> float* { return smem + GA0_OFF + idx * (BT * AS); };

    // ---- staging helpers (issue async copies; completion via wait_async_copies) ----
    auto stage_ga = [&](int kc, float* ga) {
        for (int t = tid; t < BT * KC; t += 256) {
            const int bb = t >> 6, kk = t & (KC - 1);
            const int col = gidx[n * G + kc + kk];          // x is L2-resident (64MB < 192MB)
            cp_b32(x + (size_t)(b0 + bb) * D + col, ga + bb * AS + kk);
        }
    };
    auto stage_w = [&](const float* wsrc, float* wb) {       // 64 rows x 256 cols, coalesced
        for (int t = tid; t < KC * (E / 4); t += 256) {
            const int row = t >> 6, c4 = (t & 63) << 2;
            cp_b128(wsrc + row * E + c4, wb + row * WS + c4);
        }
    };

    v8f acc[8];
#pragma unroll
    for (int t = 0; t < 8; ++t) acc[t] = (v8f)0.0f;

    // ---- WMMA compute on one staged K-chunk ----
    // alane/wlane carry ALL runtime address parts; every LDS read below is
    // base + compile-time constant -> DS immediate offsets, batchable loads.
    auto compute_chunk = [&](const float* alane, const float* wlane, v8f* a8) {
#pragma unroll
        for (int ks = 0; ks < KC; ks += 4) {
            const v2f a = *(const v2f*)(alane + ks);          // b64, imm offset
            v2f bfr[8];
#pragma unroll
            for (int t = 0; t < 8; ++t) {                     // batch all B-frag loads first
                bfr[t].x = wlane[ks * WS + t * 16];
                bfr[t].y = wlane[ks * WS + WS + t * 16];
            }
#pragma unroll
            for (int t = 0; t < 8; ++t) {
                a8[t] = __builtin_amdgcn_wmma_f32_16x16x4_f32(
                    false, a, false, bfr[t], (short)0, a8[t], false, false);
            }
        }
    };

    // ================= Layer 1: h0 = BN1(gather(x) @ W1[n] + b1[n]) =================
    const float* W1n = W1 + (size_t)n * G * E;
    stage_ga(0, ga_ptr(0));
    stage_w(W1n, wb_ptr(0));
    wait_async_copies();
    __syncthreads();
    for (int c = 0; c < G / KC; ++c) {
        const int cur = c & 1, nxt = cur ^ 1;
        if (c + 1 < G / KC) {                       // prefetch next chunk under the WMMAs
            stage_ga((c + 1) * KC, ga_ptr(nxt));
            stage_w(W1n + (size_t)(c + 1) * KC * E, wb_ptr(nxt));
        }
        compute_chunk(ga_ptr(cur) + al_off, wb_ptr(cur) + wl_off, acc);
        wait_async_copies();
        __syncthreads();
    }
    // Epilogue 1 -> h0 (overlaps dead gather buffers; all reads barriered above)
#pragma unroll
    for (int t = 0; t < 8; ++t) {
        const int e = nhalf * 128 + t * 16 + lrow;
        const float sc = gamma1[e] * rsqrtf(var1[e] + EPS);
        const float sh = (b1[n * E + e] - mean1[e]) * sc + beta1[e];
#pragma unroll
        for (int r = 0; r < 8; ++r) {
            const int row = mtile * 16 + lhi * 8 + r;   // C-layout: lanes 0-15 -> M=r, 16-31 -> M=8+r
            h0[row * HS + e] = acc[t][r] * sc + sh;
        }
    }

    // ================= Layers 2..3: h = BN2[i](h @ W2[i][n] + b2[i][n]) =================
    for (int i = 0; i < DEPTH; ++i) {
        const float* hin = (i == 0) ? h0 : h1;
        const float* W2n = W2 + ((size_t)i * NGRP + n) * E * E;
#pragma unroll
        for (int t = 0; t < 8; ++t) acc[t] = (v8f)0.0f;

        stage_w(W2n, wb_ptr(0));                    // wbuf[0] dead since layer-1 chunk c==14
        wait_async_copies();
        __syncthreads();                            // also orders h writes before reads
        for (int c = 0; c < E / KC; ++c) {
            const int cur = c & 1, nxt = cur ^ 1;
            if (c + 1 < E / KC)
                stage_w(W2n + (size_t)(c + 1) * KC * E, wb_ptr(nxt));
            compute_chunk(hin + hl_off + c * KC, wb_ptr(cur) + wl_off, acc);
            wait_async_copies();
            __syncthreads();
        }
        // Epilogue: BN, then LDS h1 (i==0) or global output (last layer)
#pragma unroll
        for (int t = 0; t < 8; ++t) {
            const int e = nhalf * 128 + t * 16 + lrow;
            const float sc = gamma2[i * E + e] * rsqrtf(var2[i * E + e] + EPS);
            const float sh = (b2[((size_t)i * NGRP + n) * E + e] - mean2[i * E + e]) * sc
                             + beta2[i * E + e];
            if (i == DEPTH - 1) {
#pragma unroll
                for (int r = 0; r < 8; ++r) {
                    const int row = mtile * 16 + lhi * 8 + r;
                    out[((size_t)(b0 + row) * NGRP + n) * E + e] = acc[t][r] * sc + sh;
                }
            } else {
#pragma unroll
                for (int r = 0; r < 8; ++r) {
                    const int row = mtile * 16 + lhi * 8 + r;
                    h1[row * HS + e] = acc[t][r] * sc + sh;
                }
            }
        }
    }
}

extern "C" void kernel_launch(void* const* d_in, const int* in_sizes, int n_in,
                              void* d_out, int out_size, void* d_ws, size_t ws_size,
                              hipStream_t stream) {
    const float* x      = (const float*)d_in[0];
    const int*   gidx   = (const int*)  d_in[1];
    const float* W1     = (const float*)d_in[2];
    const float* b1     = (const float*)d_in[3];
    const float* gamma1 = (const float*)d_in[4];
    const float* beta1  = (const float*)d_in[5];
    const float* mean1  = (const float*)d_in[6];
    const float* var1   = (const float*)d_in[7];
    const float* W2     = (const float*)d_in[8];
    const float* b2     = (const float*)d_in[9];
    const float* gamma2 = (const float*)d_in[10];
    const float* beta2  = (const float*)d_in[11];
    const float* mean2  = (const float*)d_in[12];
    const float* var2   = (const float*)d_in[13];
    float* out = (float*)d_out;

    dim3 grid(NGRP, B / BT);     // 41 x 4 workgroups
    dim3 block(256);             // 8 wave32s
    const size_t shmem = (size_t)LDS_FLOATS * sizeof(float);   // ~265 KB (< 320 KB WGP LDS)
    hipLaunchKernelGGL(locally_dense_kernel, grid, block, shmem, stream,
                       x, gidx, W1, b1, gamma1, beta1, mean1, var1,
                       W2, b2, gamma2, beta2, mean2, var2, out);
}